// FrechetMambaOperator_40269613367383
// MI455X (gfx1250) — compile-verified
//
#include <hip/hip_runtime.h>
#include <math.h>

typedef __attribute__((ext_vector_type(2))) float v2f;
typedef __attribute__((ext_vector_type(8))) float v8f;

#define L_SEQ 2048
#define D_DIM 1024
#define N_DIM 16
#define K_INIT 3072      // W*D = 3*1024
#define H0_ROWS 16384    // D*N

__device__ __forceinline__ float softplus_f(float x) {
    return (x > 20.f) ? x : log1pf(__expf(x));
}

// ---------------------------------------------------------------------------
// Kernel 1: h0[i] = dot(u[0:3072], init_w[i,:]) + init_b[i]
// Bandwidth-bound (201 MB of init_w). One wave per output row.
// ---------------------------------------------------------------------------
__global__ __launch_bounds__(256)
void h0_gemv_kernel(const float* __restrict__ u,
                    const float* __restrict__ init_w,
                    const float* __restrict__ init_b,
                    float* __restrict__ h0)
{
    int row  = (blockIdx.x * blockDim.x + threadIdx.x) >> 5;   // wave id == row
    int lane = threadIdx.x & 31;
    const float* wrow = init_w + (size_t)row * K_INIT;
    float s = 0.f;
    #pragma unroll 4
    for (int k = lane * 4; k < K_INIT; k += 128) {
        float4 a = *(const float4*)(u + k);
        float4 b = *(const float4*)(wrow + k);
        s += a.x * b.x + a.y * b.y + a.z * b.z + a.w * b.w;
    }
    #pragma unroll
    for (int off = 16; off > 0; off >>= 1) s += __shfl_xor(s, off, 32);
    if (lane == 0) h0[row] = s + init_b[row];
}

// ---------------------------------------------------------------------------
// Kernel 2: dt = softplus(u @ dt_w^T + dt_b)   [2048,1024] x [1024,1024]
// WMMA f32 16x16x4. Block = 4 waves; each wave owns one 16x16 tile of a
// 16x64 macro-tile. A-stripe staged in LDS in two 16x512 phases
// (stride 516 floats -> conflict-free ds_load_b64 across all 32 lanes).
// ---------------------------------------------------------------------------
__global__ __launch_bounds__(128)
void dt_gemm_kernel(const float* __restrict__ u,
                    const float* __restrict__ dt_w,
                    const float* __restrict__ dt_b,
                    float* __restrict__ dt_out)
{
    __shared__ float su[16 * 516];            // 33 KB, padded rows
    const int lane = threadIdx.x & 31;
    const int wave = threadIdx.x >> 5;
    const int hl   = lane >> 4;               // lane half (0/1)
    const int m    = lane & 15;
    const int row0 = blockIdx.y * 16;                 // L stripe
    const int col0 = blockIdx.x * 64 + wave * 16;     // D tile

    v8f acc = {};
    const float* bbase = dt_w + (size_t)(col0 + m) * D_DIM + 2 * hl;

    for (int phase = 0; phase < 2; ++phase) {
        const int kbase = phase * 512;
        // cooperative load of u[row0:row0+16, kbase:kbase+512] into LDS
        for (int i = threadIdx.x; i < 2048; i += 128) {   // 2048 float4s
            int r  = i >> 7;
            int c4 = (i & 127) * 4;
            *(float4*)&su[r * 516 + c4] =
                *(const float4*)(u + (size_t)(row0 + r) * D_DIM + kbase + c4);
        }
        __syncthreads();

        const float* abase = &su[m * 516 + 2 * hl];
        #pragma unroll 8
        for (int k = 0; k < 512; k += 4) {
            v2f a = *(const v2f*)(abase + k);                 // ds_load_b64
            v2f b = *(const v2f*)(bbase + kbase + k);         // global b64
            acc = __builtin_amdgcn_wmma_f32_16x16x4_f32(
                      false, a, false, b, (short)0, acc, false, false);
        }
        __syncthreads();
    }

    const float bias = dt_b[col0 + m];
    #pragma unroll
    for (int vg = 0; vg < 8; ++vg) {
        int r = row0 + vg + 8 * hl;
        dt_out[(size_t)r * D_DIM + (col0 + m)] = softplus_f(acc[vg] + bias);
    }
}

// ---------------------------------------------------------------------------
// Kernel 3: B = u @ B_w^T + B_b, C = u @ C_w^T + C_b   (N=16 wide)
// One wave per 16-row stripe; two WMMA accumulators share the A fragment.
// ---------------------------------------------------------------------------
__global__ __launch_bounds__(32)
void bc_gemm_kernel(const float* __restrict__ u,
                    const float* __restrict__ B_w, const float* __restrict__ B_b,
                    const float* __restrict__ C_w, const float* __restrict__ C_b,
                    float* __restrict__ Bout, float* __restrict__ Cout)
{
    const int lane = threadIdx.x;
    const int hl   = lane >> 4;
    const int m    = lane & 15;
    const int row0 = blockIdx.x * 16;

    v8f accB = {}, accC = {};
    const float* abase = u   + (size_t)(row0 + m) * D_DIM + 2 * hl;
    const float* bB    = B_w + (size_t)m * D_DIM + 2 * hl;
    const float* bC    = C_w + (size_t)m * D_DIM + 2 * hl;

    #pragma unroll 8
    for (int k = 0; k < D_DIM; k += 4) {
        v2f a  = *(const v2f*)(abase + k);
        v2f b1 = *(const v2f*)(bB + k);
        v2f b2 = *(const v2f*)(bC + k);
        accB = __builtin_amdgcn_wmma_f32_16x16x4_f32(
                   false, a, false, b1, (short)0, accB, false, false);
        accC = __builtin_amdgcn_wmma_f32_16x16x4_f32(
                   false, a, false, b2, (short)0, accC, false, false);
    }

    const float biasB = B_b[m], biasC = C_b[m];
    #pragma unroll
    for (int vg = 0; vg < 8; ++vg) {
        int r = row0 + vg + 8 * hl;
        Bout[r * N_DIM + m] = accB[vg] + biasB;
        Cout[r * N_DIM + m] = accC[vg] + biasC;
    }
}

// ---------------------------------------------------------------------------
// Kernel 4: selective scan. Thread = one (d,n) state element; h in register.
// 16-lane __shfl_xor tree reduces sum_n C*h; lane n==0 writes softplus output.
// ---------------------------------------------------------------------------
__global__ __launch_bounds__(256)
void scan_kernel(const float* __restrict__ dt, const float* __restrict__ Bm,
                 const float* __restrict__ Cm, const float* __restrict__ u,
                 const float* __restrict__ A_log, const float* __restrict__ D_w,
                 const float* __restrict__ scale_p, const float* __restrict__ h0,
                 float* __restrict__ out)
{
    const int t = blockIdx.x * 256 + threadIdx.x;  // 0..16383
    const int d = t >> 4;
    const int n = t & 15;

    const float A  = -__expf(A_log[n]);
    const float Dw = D_w[d];
    const float sc = fmaxf(scale_p[0], 0.5f);
    float h = h0[t];

    for (int l = 0; l < L_SEQ; ++l) {
        float dtv = dt[(size_t)l * D_DIM + d];
        float uv  = u [(size_t)l * D_DIM + d];
        float Bv  = Bm[l * N_DIM + n];
        float Cv  = Cm[l * N_DIM + n];

        float dtA = dtv * A;
        float dA  = __expf(dtA);
        float dB  = expm1f(dtA) / (dtA + 1e-9f) * Bv;   // ZOH discretization
        h = fmaf(dA, h, dB * uv);

        float yc = Cv * h;
        yc += __shfl_xor(yc, 1, 32);
        yc += __shfl_xor(yc, 2, 32);
        yc += __shfl_xor(yc, 4, 32);
        yc += __shfl_xor(yc, 8, 32);

        if (n == 0) {
            float y = yc + Dw * uv;
            out[(size_t)l * D_DIM + d] = softplus_f(y) * sc;
        }
    }
}

// ---------------------------------------------------------------------------
extern "C" void kernel_launch(void* const* d_in, const int* in_sizes, int n_in,
                              void* d_out, int out_size, void* d_ws, size_t ws_size,
                              hipStream_t stream) {
    const float* u_seq  = (const float*)d_in[0];
    const float* A_log  = (const float*)d_in[1];
    const float* dt_w   = (const float*)d_in[2];
    const float* dt_b   = (const float*)d_in[3];
    const float* B_w    = (const float*)d_in[4];
    const float* B_b    = (const float*)d_in[5];
    const float* C_w    = (const float*)d_in[6];
    const float* C_b    = (const float*)d_in[7];
    const float* D_w    = (const float*)d_in[8];
    const float* scale  = (const float*)d_in[9];
    const float* init_w = (const float*)d_in[10];
    const float* init_b = (const float*)d_in[11];
    float* out = (float*)d_out;

    // workspace layout (floats): h0[16384] | B[32768] | C[32768] | dt[2M]
    float* ws    = (float*)d_ws;
    float* h0    = ws;
    float* Bmat  = ws + 16384;
    float* Cmat  = ws + 16384 + 32768;
    float* dtmat = ws + 16384 + 2 * 32768;

    // 1) h0 GEMV: 16384 rows, 1 wave each, 8 waves/block
    h0_gemv_kernel<<<H0_ROWS / 8, 256, 0, stream>>>(u_seq, init_w, init_b, h0);

    // 2) dt GEMM (WMMA f32): grid (D/64, L/16), 4 waves/block
    dt_gemm_kernel<<<dim3(D_DIM / 64, L_SEQ / 16), 128, 0, stream>>>(
        u_seq, dt_w, dt_b, dtmat);

    // 3) B/C skinny GEMMs (WMMA f32): one wave per 16-row stripe
    bc_gemm_kernel<<<L_SEQ / 16, 32, 0, stream>>>(
        u_seq, B_w, B_b, C_w, C_b, Bmat, Cmat);

    // 4) sequential scan, parallel over D*N = 16384 threads
    scan_kernel<<<H0_ROWS / 256, 256, 0, stream>>>(
        dtmat, Bmat, Cmat, u_seq, A_log, D_w, scale, h0, out);
}